// PostProcess_66331474919775
// MI455X (gfx1250) — compile-verified
//
#include <hip/hip_runtime.h>
#include <hip/hip_bf16.h>

#define BDIM 4
#define ADIM 8732
#define CDIM 81
#define NCLS 80
#define NCAND (ADIM * NCLS)   /* 698560 */
#define MAXOUT 100
#define SCORETHR 0.05f
#define MINSZ (1.0f / 300.0f)
#define CRIT 0.5f
#define NEGINF (-__builtin_inff())
#define SEGSZ 256
#define NSEG ((ADIM + SEGSZ - 1) / SEGSZ)   /* 35 */
#define NSEGTOT (NCLS * NSEG)               /* 2800 */

// ---------------------------------------------------------------------------
// Kernel 1: decode + clip + size check + softmax + score thresholding.
// boxes_ws : [B][A][4] clipped boxes
// s_ws     : [B][80][A] scores TRANSPOSED (prob if valid else -inf),
//            row r = class-1, column a = anchor. Coalesced per-class stores.
// ---------------------------------------------------------------------------
__global__ void ssd_prep_kernel(const float* __restrict__ ploc,
                                const float* __restrict__ plabel,
                                const float* __restrict__ ancs,
                                float* __restrict__ boxes_ws,
                                float* __restrict__ s_ws)
{
    int t = blockIdx.x * blockDim.x + threadIdx.x;
    if (t >= BDIM * ADIM) return;
    int b = t / ADIM;
    int a = t - b * ADIM;

    // ---- decode ----
    float ax = ancs[a * 4 + 0], ay = ancs[a * 4 + 1];
    float aw = ancs[a * 4 + 2], ah = ancs[a * 4 + 3];
    float px = ploc[((size_t)b * 4 + 0) * ADIM + a];
    float py = ploc[((size_t)b * 4 + 1) * ADIM + a];
    float pw = ploc[((size_t)b * 4 + 2) * ADIM + a];
    float ph = ploc[((size_t)b * 4 + 3) * ADIM + a];
    float cx = 0.1f * px * aw + ax;
    float cy = 0.1f * py * ah + ay;
    float w  = __expf(0.2f * pw) * aw;
    float h  = __expf(0.2f * ph) * ah;
    float x1 = fminf(fmaxf(cx - 0.5f * w, 0.f), 1.f);
    float y1 = fminf(fmaxf(cy - 0.5f * h, 0.f), 1.f);
    float x2 = fminf(fmaxf(cx + 0.5f * w, 0.f), 1.f);
    float y2 = fminf(fmaxf(cy + 0.5f * h, 0.f), 1.f);
    bool size_ok = ((x2 - x1) >= MINSZ) && ((y2 - y1) >= MINSZ);

    size_t bo = ((size_t)b * ADIM + a) * 4;
    boxes_ws[bo + 0] = x1;
    boxes_ws[bo + 1] = y1;
    boxes_ws[bo + 2] = x2;
    boxes_ws[bo + 3] = y2;

    // ---- softmax over 81 classes (three streaming passes; L2-resident) ----
    const float* pl = plabel + (size_t)b * CDIM * ADIM + a;
    float m = NEGINF;
    for (int c = 0; c < CDIM; ++c)
        m = fmaxf(m, pl[(size_t)c * ADIM]);
    float sum = 0.f;
    for (int c = 0; c < CDIM; ++c)
        sum += __expf(pl[(size_t)c * ADIM] - m);
    float inv = 1.0f / sum;

    float* sbase = s_ws + (size_t)b * NCAND + a;   // column a of image b
    for (int c = 1; c < CDIM; ++c) {
        float p = __expf(pl[(size_t)c * ADIM] - m) * inv;
        sbase[(size_t)(c - 1) * ADIM] = (size_ok && p > SCORETHR) ? p : NEGINF;
    }
}

// ---------------------------------------------------------------------------
// Wave-level argmax reduce on (value, packed-code); higher value wins,
// ties prefer lower code (code = row*ADIM + anchor, -1 = empty).
// ---------------------------------------------------------------------------
__device__ __forceinline__ void wave_argmax(float& v, int& c)
{
    for (int o = 16; o > 0; o >>= 1) {
        float ov = __shfl_down(v, o, 32);
        int   oc = __shfl_down(c, o, 32);
        if (ov > v || (ov == v && oc >= 0 && (c < 0 || oc < c))) { v = ov; c = oc; }
    }
}

// ---------------------------------------------------------------------------
// Kernel 2: per-image greedy NMS. One 1024-thread (32-wave) workgroup/image.
//  - boxes (139.7 KB) staged into LDS once via the Tensor Data Mover
//  - hierarchical argmax: per-segment maxes (80 rows x 35 segs) kept in LDS
//  - cross-label IoU is provably zero (offset >= 2 vs box extent <= 1), so
//    suppression touches only the winner's label row; the suppress pass and
//    the segment-max rebuild for that row are fused.
// ---------------------------------------------------------------------------
typedef unsigned int u32x4 __attribute__((ext_vector_type(4)));
typedef int          i32x8 __attribute__((ext_vector_type(8)));
typedef int          i32x4 __attribute__((ext_vector_type(4)));

__global__ void __launch_bounds__(1024)
ssd_nms_kernel(const float* __restrict__ boxes_ws,
               float* __restrict__ s_ws,
               float* __restrict__ ob,   // [B][100][4]
               int*   __restrict__ ol,   // [B][100]
               float* __restrict__ os)   // [B][100]
{
    extern __shared__ float smem[];
    float4* lbox = (float4*)smem;                   // ADIM boxes
    float*  segv = smem + ADIM * 4;                 // NSEGTOT segment maxes
    int*    segc = (int*)(segv + NSEGTOT);          // NSEGTOT packed codes
    float*  rv   = (float*)(segc + NSEGTOT);        // 32 per-wave partials
    int*    ri   = (int*)(rv + 32);                 // 32 per-wave codes
    float*  bc   = (float*)(ri + 32);               // [0..3] box, [4] area
    int*    bcI  = (int*)bc;                        // [5] alive, [6] lab, [7] anc

    const int b    = blockIdx.x;
    const int tid  = threadIdx.x;
    const int wid  = tid >> 5;
    const int lane = tid & 31;

    float* sb = s_ws + (size_t)b * NCAND;           // [80][ADIM]
    const float* gbox = boxes_ws + (size_t)b * ADIM * 4;

    // ---- stage this image's boxes into LDS via the Tensor Data Mover ----
#if __has_builtin(__builtin_amdgcn_tensor_load_to_lds) && __has_builtin(__builtin_amdgcn_s_wait_tensorcnt)
    if (wid == 0) {
        const unsigned int ndw = ADIM * 4;                  // 34928 dwords, one row
        unsigned long long ga = (unsigned long long)(uintptr_t)gbox;
        unsigned int laddr = (unsigned int)(uintptr_t)lbox; // low 32 bits = LDS offset

        u32x4 g0;
        g0[0] = 1u;                                         // count=1, user mode
        g0[1] = laddr;                                      // lds_addr
        g0[2] = (unsigned int)(ga & 0xFFFFFFFFull);         // global_addr[31:0]
        g0[3] = (unsigned int)((ga >> 32) & 0x01FFFFFFull)  // global_addr[56:32]
              | 0x80000000u;                                // type = 2 ("image")

        i32x8 g1;
        g1[0] = (int)(2u << 16);                            // data_size = 4B, mask=0
        g1[1] = (int)((ndw & 0xFFFFu) << 16);               // tensor_dim0[15:0]
        g1[2] = (int)(((ndw >> 16) & 0xFFFFu)               // tensor_dim0[31:16]
              | (1u << 16));                                // tensor_dim1 = 1
        g1[3] = (int)((ndw & 0xFFFFu) << 16);               // tile_dim0 = ndw
        g1[4] = 1;                                          // tile_dim1 = 1
        g1[5] = (int)ndw;                                   // tensor_dim0_stride
        g1[6] = 0;
        g1[7] = 0;

        i32x4 gz4 = {0, 0, 0, 0};
        i32x8 gz8 = {0, 0, 0, 0, 0, 0, 0, 0};
        __builtin_amdgcn_tensor_load_to_lds(g0, g1, gz4, gz4, gz8, 0);
        __builtin_amdgcn_s_wait_tensorcnt(0);
    }
#else
    for (int i = tid; i < ADIM * 4; i += blockDim.x)
        smem[i] = gbox[i];
#endif

    // ---- initial segment-max build (one wave per segment, coalesced) ----
    for (int seg = wid; seg < NSEGTOT; seg += 32) {
        int r  = seg / NSEG;
        int s  = seg - r * NSEG;
        int a0 = s * SEGSZ;
        const float* row = sb + (size_t)r * ADIM;
        float mv = NEGINF;
        int   mc = -1;
        for (int a = a0 + lane; a < a0 + SEGSZ && a < ADIM; a += 32) {
            float v = row[a];
            if (v > mv) { mv = v; mc = r * ADIM + a; }
        }
        wave_argmax(mv, mc);
        if (lane == 0) { segv[seg] = mv; segc[seg] = mc; }
    }
    __syncthreads();

    // ---- 100 sequential greedy-NMS steps ----
    for (int it = 0; it < MAXOUT; ++it) {
        // argmax over 2800 LDS segment maxes
        float bestv = NEGINF;
        int   bestc = -1;
        for (int k = tid; k < NSEGTOT; k += 1024) {
            float v = segv[k];
            if (v > bestv || (v == bestv && bestc >= 0 && segc[k] >= 0 &&
                              segc[k] < bestc)) { bestv = v; bestc = segc[k]; }
        }
        wave_argmax(bestv, bestc);
        if (lane == 0) { rv[wid] = bestv; ri[wid] = bestc; }
        __syncthreads();
        if (wid == 0) {
            bestv = rv[lane];
            bestc = ri[lane];
            wave_argmax(bestv, bestc);
            if (lane == 0) {
                bool alive = (bestv > NEGINF) && (bestc >= 0);
                float4 bx = make_float4(0.f, 0.f, 0.f, 0.f);
                int lab = 0, anc = 0;
                if (alive) {
                    lab = bestc / ADIM + 1;
                    anc = bestc - (lab - 1) * ADIM;
                    bx  = lbox[anc];
                }
                size_t oidx = (size_t)b * MAXOUT + it;
                ob[oidx * 4 + 0] = alive ? bx.x : 0.f;
                ob[oidx * 4 + 1] = alive ? bx.y : 0.f;
                ob[oidx * 4 + 2] = alive ? bx.z : 0.f;
                ob[oidx * 4 + 3] = alive ? bx.w : 0.f;
                ol[oidx] = alive ? lab : 0;
                os[oidx] = alive ? bestv : 0.f;

                bc[0] = bx.x; bc[1] = bx.y; bc[2] = bx.z; bc[3] = bx.w;
                bc[4] = (bx.z - bx.x) * (bx.w - bx.y);   // area (offsets cancel)
                bcI[5] = alive ? 1 : 0;
                bcI[6] = lab;
                bcI[7] = anc;
            }
        }
        __syncthreads();

        // fused suppression + segment-max rebuild of the winner's label row
        if (bcI[5]) {
            const int r    = bcI[6] - 1;
            const int banc = bcI[7];
            const float bx1 = bc[0], by1 = bc[1], bx2 = bc[2], by2 = bc[3];
            const float ba  = bc[4];
            float* row = sb + (size_t)r * ADIM;
            for (int seg = wid; seg < NSEG; seg += 32) {
                int a0 = seg * SEGSZ;
                float mv = NEGINF;
                int   mc = -1;
                for (int a = a0 + lane; a < a0 + SEGSZ && a < ADIM; a += 32) {
                    float v = row[a];
                    float nv = v;
                    if (v > NEGINF) {
                        if (a == banc) {
                            nv = NEGINF;                    // .at[idx].set(-inf)
                        } else {
                            float4 c = lbox[a];             // same label: offsets cancel
                            float iw = fmaxf(fminf(bx2, c.z) - fmaxf(bx1, c.x), 0.f);
                            float ih = fmaxf(fminf(by2, c.w) - fmaxf(by1, c.y), 0.f);
                            float inter = iw * ih;
                            float ca = (c.z - c.x) * (c.w - c.y);
                            float iou = inter / (ca + ba - inter + 1e-12f);
                            if (iou > CRIT) nv = NEGINF;
                        }
                        if (nv != v) row[a] = nv;
                    }
                    if (nv > mv) { mv = nv; mc = r * ADIM + a; }
                }
                wave_argmax(mv, mc);
                if (lane == 0) { segv[r * NSEG + seg] = mv; segc[r * NSEG + seg] = mc; }
            }
        }
        __syncthreads();
    }
}

// ---------------------------------------------------------------------------
extern "C" void kernel_launch(void* const* d_in, const int* in_sizes, int n_in,
                              void* d_out, int out_size, void* d_ws, size_t ws_size,
                              hipStream_t stream)
{
    (void)in_sizes; (void)n_in; (void)out_size; (void)ws_size;
    const float* ploc   = (const float*)d_in[0];
    const float* plabel = (const float*)d_in[1];
    const float* ancs   = (const float*)d_in[2];

    float* boxes_ws = (float*)d_ws;                       // B*A*4 floats
    float* s_ws     = boxes_ws + (size_t)BDIM * ADIM * 4; // B*80*A floats

    float* outf = (float*)d_out;
    float* ob = outf;                                     // B*100*4
    int*   ol = (int*)(outf + (size_t)BDIM * MAXOUT * 4); // B*100 (int32)
    float* os = outf + (size_t)BDIM * MAXOUT * 4 + (size_t)BDIM * MAXOUT;

    int total = BDIM * ADIM;
    int blk = 256;
    ssd_prep_kernel<<<(total + blk - 1) / blk, blk, 0, stream>>>(
        ploc, plabel, ancs, boxes_ws, s_ws);

    // LDS: boxes (ADIM*16B) + segment tree (2*NSEGTOT*4B) + scratch
    size_t lds_bytes = ((size_t)ADIM * 4 + 2 * NSEGTOT + 32 + 32 + 8) * sizeof(float);
    ssd_nms_kernel<<<BDIM, 1024, lds_bytes, stream>>>(
        boxes_ws, s_ws, ob, ol, os);
}